// RPNInference_82394652607038
// MI455X (gfx1250) — compile-verified
//
#include <hip/hip_runtime.h>

// ---------------- problem constants (match reference) ----------------
#define NN 16
#define AA 3
#define HH 256
#define WW 384
#define MM (AA * HH * WW)          // 294912 anchors per image
#define KCAND 6000                 // PRE_NMS_TOP_N
#define NOUT 1000                  // POST_NMS_TOP_N
#define NMS_T 0.7f
#define NEGV (-1.0e10f)
#define XFORM_CLIP 4.135166556742356f   // log(1000/16)
#define IMGW_M1 1535.0f
#define IMGH_M1 1023.0f

#define HBLK 64                    // blocks per image for histogram pass
#define CBLK 256
#define NBLK (MM / CBLK)           // 1152 one-element-per-thread blocks

// ---------------- CDNA5 async global->LDS path (feature-detected) ----
#if defined(__gfx1250__) && __has_builtin(__builtin_amdgcn_global_load_async_to_lds_b128) && __has_builtin(__builtin_amdgcn_s_wait_asynccnt)
#define HAVE_ASYNC_LDS 1
#else
#define HAVE_ASYNC_LDS 0
#endif

#if HAVE_ASYNC_LDS
typedef int v4i_t __attribute__((ext_vector_type(4)));
typedef __attribute__((address_space(1))) v4i_t* g_v4i_ptr;
typedef __attribute__((address_space(3))) v4i_t* l_v4i_ptr;

__device__ __forceinline__ void async_cp16(void* lds, const void* g) {
  // b128 async copy: per-lane 16B global -> LDS, tracked by ASYNCcnt
  __builtin_amdgcn_global_load_async_to_lds_b128(
      (g_v4i_ptr)g, (l_v4i_ptr)lds, /*offset=*/0, /*cpol=*/0);
}
#endif

// float -> monotone-ascending unsigned key
__device__ __forceinline__ unsigned f2k(float f) {
  unsigned u = __float_as_uint(f);
  return (u & 0x80000000u) ? ~u : (u | 0x80000000u);
}

// ---------------- k_init: zero histograms + counters ------------------
__global__ void k_init(unsigned* hist, unsigned* cnt) {
  int i = blockIdx.x * blockDim.x + threadIdx.x;
  if (i < NN * 4 * 256) hist[i] = 0u;
  if (i < NN) cnt[i] = 0u;
}

// ---------------- radix-select pass: 256-bin histogram ----------------
__global__ void k_hist(const float* __restrict__ obj, const uint2* __restrict__ state,
                       unsigned* __restrict__ hist, int level) {
  __shared__ unsigned lh[256];
  const int n = blockIdx.y;
  lh[threadIdx.x] = 0u;            // blockDim.x == 256
  __syncthreads();

  unsigned prefix = 0u, mask = 0u;
  if (level > 0) {
    prefix = state[n].x;           // high 8*level bits fixed, rest zero
    mask = 0xFFFFFFFFu << (32 - 8 * level);
  }
  const int shift = 24 - 8 * level;
  const float* o = obj + (size_t)n * MM;
  const int per = MM / gridDim.x;  // 4608
  const int s0 = blockIdx.x * per;
  for (int i = s0 + threadIdx.x; i < s0 + per; i += 256) {
    unsigned key = f2k(o[i]);
    if (level == 0 || (key & mask) == prefix)
      atomicAdd(&lh[(key >> shift) & 255u], 1u);
  }
  __syncthreads();
  atomicAdd(&hist[((size_t)n * 4 + level) * 256 + threadIdx.x], lh[threadIdx.x]);
}

// ---------------- radix-select pass: pick bin, update state -----------
__global__ void k_sel(uint2* __restrict__ state, const unsigned* __restrict__ hist, int level) {
  const int n = blockIdx.x;
  if (threadIdx.x != 0) return;
  unsigned krem = (level == 0) ? (unsigned)KCAND : state[n].y;
  unsigned prefix = (level == 0) ? 0u : state[n].x;
  const unsigned* h = hist + ((size_t)n * 4 + level) * 256;
  unsigned cum = 0u;
  int b = 255;
  for (; b >= 0; --b) {
    unsigned c = h[b];
    if (cum + c >= krem) { krem -= cum; break; }
    cum += c;
  }
  if (b < 0) b = 0;
  prefix |= ((unsigned)b) << (24 - 8 * level);
  state[n].x = prefix;             // after level 3: exact threshold key T
  state[n].y = krem;               // after level 3: #(==T) to take, index order
}

// ---------------- boundary (==T) counts per block ---------------------
__global__ void k_bcount(const float* __restrict__ obj, const uint2* __restrict__ state,
                         unsigned* __restrict__ bcnt) {
  const int n = blockIdx.y;
  const int l = blockIdx.x * CBLK + threadIdx.x;
  const unsigned T = state[n].x;
  const bool bd = (f2k(obj[(size_t)n * MM + l]) == T);
  const int lane = threadIdx.x & 31, wid = threadIdx.x >> 5;
  unsigned long long mb = __ballot(bd);
  __shared__ unsigned wc[8];
  if (lane == 0) wc[wid] = __popc((unsigned)mb);
  __syncthreads();
  if (threadIdx.x == 0) {
    unsigned s = 0;
    for (int i = 0; i < 8; ++i) s += wc[i];
    bcnt[(size_t)n * NBLK + blockIdx.x] = s;
  }
}

// ---------------- exclusive prefix of boundary counts -----------------
__global__ void k_bscan(const unsigned* __restrict__ bcnt, unsigned* __restrict__ bpref) {
  const int n = blockIdx.x;
  if (threadIdx.x != 0) return;
  unsigned run = 0;
  for (int i = 0; i < NBLK; ++i) {
    bpref[(size_t)n * NBLK + i] = run;
    run += bcnt[(size_t)n * NBLK + i];
  }
}

// ---------------- compact: decode+clip the exact top-K ----------------
__global__ void k_compact(const float* __restrict__ anchors, const float* __restrict__ obj,
                          const float* __restrict__ reg, const uint2* __restrict__ state,
                          const unsigned* __restrict__ bpref, unsigned* __restrict__ cnt,
                          float* __restrict__ cx1, float* __restrict__ cy1,
                          float* __restrict__ cx2, float* __restrict__ cy2,
                          float* __restrict__ csc) {
  const int n = blockIdx.y;
  const int l = blockIdx.x * CBLK + threadIdx.x;
  const float logit = obj[(size_t)n * MM + l];
  const unsigned key = f2k(logit);
  const uint2 st = state[n];
  const unsigned T = st.x, krem = st.y;

  // ballot-prefix rank for ==T elements (all threads participate)
  const bool bd = (key == T);
  const int lane = threadIdx.x & 31, wid = threadIdx.x >> 5;
  unsigned long long mb = __ballot(bd);
  const unsigned lmask = (1u << lane) - 1u;
  const unsigned wrank = __popc((unsigned)mb & lmask);
  __shared__ unsigned wc[8];
  if (lane == 0) wc[wid] = __popc((unsigned)mb);
  __syncthreads();
  unsigned wpref = 0;
  for (int i = 0; i < wid; ++i) wpref += wc[i];

  int pos = -1;
  if (key > T) {
    pos = (int)atomicAdd(&cnt[n], 1u);                 // [0, K-krem)
  } else if (bd) {
    unsigned grank = bpref[(size_t)n * NBLK + blockIdx.x] + wpref + wrank;
    if (grank < krem) pos = (int)(KCAND - krem + grank);   // [K-krem, K)
  }
  if (pos < 0) return;

  // decode (reference _decode) + clip
  const int a = l / (HH * WW);
  const int rem = l % (HH * WW);
  const int h = rem / WW, w = rem % WW;
  const int m = (h * WW + w) * AA + a;
  const float ax1 = anchors[m * 4 + 0], ay1 = anchors[m * 4 + 1];
  const float ax2 = anchors[m * 4 + 2], ay2 = anchors[m * 4 + 3];
  const float aw = ax2 - ax1 + 1.0f, ah = ay2 - ay1 + 1.0f;
  const float acx = ax1 + 0.5f * aw, acy = ay1 + 0.5f * ah;
  const size_t rb = (size_t)n * (AA * 4) * HH * WW;
  const float dx = reg[rb + ((a * 4 + 0) * HH + h) * (size_t)WW + w];
  const float dy = reg[rb + ((a * 4 + 1) * HH + h) * (size_t)WW + w];
  const float dw = fminf(reg[rb + ((a * 4 + 2) * HH + h) * (size_t)WW + w], XFORM_CLIP);
  const float dh = fminf(reg[rb + ((a * 4 + 3) * HH + h) * (size_t)WW + w], XFORM_CLIP);
  const float pcx = dx * aw + acx, pcy = dy * ah + acy;
  const float pw = expf(dw) * aw, ph = expf(dh) * ah;
  float x1 = pcx - 0.5f * pw, y1 = pcy - 0.5f * ph;
  float x2 = pcx + 0.5f * pw - 1.0f, y2 = pcy + 0.5f * ph - 1.0f;
  x1 = fminf(fmaxf(x1, 0.0f), IMGW_M1);
  x2 = fminf(fmaxf(x2, 0.0f), IMGW_M1);
  y1 = fminf(fmaxf(y1, 0.0f), IMGH_M1);
  y2 = fminf(fmaxf(y2, 0.0f), IMGH_M1);
  // MIN_SIZE==0 filter is a no-op: pred_w,pred_h>0 => clipped ws,hs >= 0.
  const size_t o = (size_t)n * KCAND + pos;
  cx1[o] = x1; cy1[o] = y1; cx2[o] = x2; cy2[o] = y2;
  csc[o] = 1.0f / (1.0f + expf(-logit));               // sigmoid
}

// ---------------- NMS: 1 workgroup per image, all state in LDS --------
__global__ void __launch_bounds__(1024, 1)
k_nms(const float* __restrict__ cx1, const float* __restrict__ cy1,
      const float* __restrict__ cx2, const float* __restrict__ cy2,
      const float* __restrict__ csc, float* __restrict__ out) {
  __shared__ float s_x1[KCAND], s_y1[KCAND], s_x2[KCAND], s_y2[KCAND];
  __shared__ float s_sc[KCAND], s_ar[KCAND];             // 144 KB total (CDNA5 WGP LDS)
  __shared__ float s_wv[32];
  __shared__ int s_wi[32];
  __shared__ float s_sel[5];

  const int n = blockIdx.x;
  const int tid = threadIdx.x;
  const int lane = tid & 31, wid = tid >> 5;
  const size_t base = (size_t)n * KCAND;

#if HAVE_ASYNC_LDS
  // async global->LDS staging of the candidate set, b128 per lane
  for (int i = tid * 4; i < KCAND; i += 1024 * 4) {
    async_cp16(&s_x1[i], cx1 + base + i);
    async_cp16(&s_y1[i], cy1 + base + i);
    async_cp16(&s_x2[i], cx2 + base + i);
    async_cp16(&s_y2[i], cy2 + base + i);
    async_cp16(&s_sc[i], csc + base + i);
  }
  __builtin_amdgcn_s_wait_asynccnt(0);
  __syncthreads();
#else
  for (int i = tid; i < KCAND; i += 1024) {
    s_x1[i] = cx1[base + i]; s_y1[i] = cy1[base + i];
    s_x2[i] = cx2[base + i]; s_y2[i] = cy2[base + i];
    s_sc[i] = csc[base + i];
  }
  __syncthreads();
#endif
  for (int i = tid; i < KCAND; i += 1024)
    s_ar[i] = (s_x2[i] - s_x1[i] + 1.0f) * (s_y2[i] - s_y1[i] + 1.0f);
  __syncthreads();

  for (int it = 0; it < NOUT; ++it) {
    // per-thread argmax (first-max tie rule via strict >, then smaller index)
    float bv = -3.0e38f;
    int bi = 0;
    for (int i = tid; i < KCAND; i += 1024) {
      float v = s_sc[i];
      if (v > bv) { bv = v; bi = i; }
    }
    // wave32 reduction
    for (int off = 16; off > 0; off >>= 1) {
      float ov = __shfl_xor(bv, off, 32);
      int oi = __shfl_xor(bi, off, 32);
      if (ov > bv || (ov == bv && oi < bi)) { bv = ov; bi = oi; }
    }
    if (lane == 0) { s_wv[wid] = bv; s_wi[wid] = bi; }
    __syncthreads();
    if (wid == 0) {
      bv = s_wv[lane]; bi = s_wi[lane];                  // exactly 32 waves
      for (int off = 16; off > 0; off >>= 1) {
        float ov = __shfl_xor(bv, off, 32);
        int oi = __shfl_xor(bi, off, 32);
        if (ov > bv || (ov == bv && oi < bi)) { bv = ov; bi = oi; }
      }
      if (lane == 0) {
        s_sel[0] = s_x1[bi]; s_sel[1] = s_y1[bi];
        s_sel[2] = s_x2[bi]; s_sel[3] = s_y2[bi];
        s_sel[4] = bv;
      }
    }
    __syncthreads();

    const float bsc = s_sel[4];
    if (bsc <= 0.5f * NEGV) {
      // exhausted: all remaining outputs are zeros / NEG, fill and exit
      for (int q = it * 5 + tid; q < NOUT * 5; q += 1024)
        out[(size_t)n * NOUT * 5 + q] = ((q % 5) == 4) ? NEGV : 0.0f;
      return;
    }
    const float bx1 = s_sel[0], by1 = s_sel[1], bx2 = s_sel[2], by2 = s_sel[3];
    if (tid == 0) {
      float* o = out + ((size_t)n * NOUT + it) * 5;
      o[0] = bx1; o[1] = by1; o[2] = bx2; o[3] = by2; o[4] = bsc;
    }
    const float barea = (bx2 - bx1 + 1.0f) * (by2 - by1 + 1.0f);
    for (int i = tid; i < KCAND; i += 1024) {
      float xx1 = fmaxf(bx1, s_x1[i]);
      float yy1 = fmaxf(by1, s_y1[i]);
      float xx2 = fminf(bx2, s_x2[i]);
      float yy2 = fminf(by2, s_y2[i]);
      float inter = fmaxf(xx2 - xx1 + 1.0f, 0.0f) * fmaxf(yy2 - yy1 + 1.0f, 0.0f);
      float iou = inter / (s_ar[i] + barea - inter);
      if (iou > NMS_T) s_sc[i] = NEGV;                   // suppresses selected too (iou=1)
    }
    __syncthreads();
  }
}

// ---------------- launcher -------------------------------------------
static inline size_t al256(size_t x) { return (x + 255) & ~(size_t)255; }

extern "C" void kernel_launch(void* const* d_in, const int* in_sizes, int n_in,
                              void* d_out, int out_size, void* d_ws, size_t ws_size,
                              hipStream_t stream) {
  const float* anchors = (const float*)d_in[0];   // [M,4]
  const float* obj     = (const float*)d_in[1];   // [N,A,H,W]
  const float* reg     = (const float*)d_in[2];   // [N,A*4,H,W]
  float* out = (float*)d_out;                     // [N,1000,5]

  // workspace carve-up (256B-aligned chunks; ~2.2 MB total)
  char* w = (char*)d_ws;
  size_t off = 0;
  unsigned* hist  = (unsigned*)(w + off); off = al256(off + (size_t)NN * 4 * 256 * 4);
  uint2*    state = (uint2*)   (w + off); off = al256(off + (size_t)NN * 8);
  unsigned* cnt   = (unsigned*)(w + off); off = al256(off + (size_t)NN * 4);
  unsigned* bcnt  = (unsigned*)(w + off); off = al256(off + (size_t)NN * NBLK * 4);
  unsigned* bpref = (unsigned*)(w + off); off = al256(off + (size_t)NN * NBLK * 4);
  float* cx1 = (float*)(w + off); off = al256(off + (size_t)NN * KCAND * 4);
  float* cy1 = (float*)(w + off); off = al256(off + (size_t)NN * KCAND * 4);
  float* cx2 = (float*)(w + off); off = al256(off + (size_t)NN * KCAND * 4);
  float* cy2 = (float*)(w + off); off = al256(off + (size_t)NN * KCAND * 4);
  float* csc = (float*)(w + off); off = al256(off + (size_t)NN * KCAND * 4);

  k_init<<<dim3((NN * 4 * 256 + 255) / 256), 256, 0, stream>>>(hist, cnt);

  for (int level = 0; level < 4; ++level) {
    k_hist<<<dim3(HBLK, NN), 256, 0, stream>>>(obj, state, hist, level);
    k_sel<<<dim3(NN), 32, 0, stream>>>(state, hist, level);
  }
  k_bcount<<<dim3(NBLK, NN), CBLK, 0, stream>>>(obj, state, bcnt);
  k_bscan<<<dim3(NN), 32, 0, stream>>>(bcnt, bpref);
  k_compact<<<dim3(NBLK, NN), CBLK, 0, stream>>>(anchors, obj, reg, state, bpref, cnt,
                                                 cx1, cy1, cx2, cy2, csc);
  k_nms<<<dim3(NN), 1024, 0, stream>>>(cx1, cy1, cx2, cy2, csc, out);
}